// _TopKRouter_37409165148804
// MI455X (gfx1250) — compile-verified
//
#include <hip/hip_runtime.h>
#include <hip/hip_bf16.h>
#include <math.h>
#include <stdint.h>

// ---------------- problem constants (from reference) ----------------
#define T_TOK    16384      // 4 * 4096 tokens
#define DM       2048       // d_model
#define NEXP     64         // num experts
#define CAP      512        // ceil(T/E * 2.0)
#define ALPHA_C  0.01f

// ---------------- tiling ----------------
#define TOK_PER_BLK 64                    // tokens per phaseA/phaseC block
#define NBLK        (T_TOK / TOK_PER_BLK) // 256
#define KC          128                   // K-slab staged in LDS
#define SW          132                   // padded LDS stride for W slab (floats) -> conflict-free b64 reads
#define SL          66                    // padded LDS stride for logits (floats)
#define SMEM_FLOATS 8448                  // max(64*SW, 64*SL) == 8448 for both (overlaid)

typedef __attribute__((ext_vector_type(2))) float v2f;
typedef __attribute__((ext_vector_type(8))) float v8f;

// ---------------- workspace layout (bytes) ----------------
#define WS_CNT   0                            // int   [NBLK*NEXP]
#define WS_OFFS  (WS_CNT  + NBLK*NEXP*4)      // int   [NBLK*NEXP]
#define WS_PSUM  (WS_OFFS + NBLK*NEXP*4)      // float [NBLK*NEXP]
#define WS_LSE2  (WS_PSUM + NBLK*NEXP*4)      // float [NBLK]
#define WS_TOPI  (WS_LSE2 + NBLK*4)           // int2  [T_TOK]
#define WS_TOPP  (WS_TOPI + T_TOK*8)          // float2[T_TOK]

// =====================================================================
// Phase A: WMMA GEMM (x @ W^T + b) -> softmax -> top2 -> per-block stats
// =====================================================================
__global__ __launch_bounds__(128)
void moe_phaseA(const float* __restrict__ x, const float* __restrict__ Wg,
                const float* __restrict__ bg,
                int*   __restrict__ cnt_blk,  float* __restrict__ psum_blk,
                float* __restrict__ lse2_blk,
                int2*  __restrict__ topi,     float2* __restrict__ topp)
{
    __shared__ float smem[SMEM_FLOATS];       // W slab during GEMM, logits after
    __shared__ float mx_s[TOK_PER_BLK];
    __shared__ float inv_s[TOK_PER_BLK];
    __shared__ float lse2_s[TOK_PER_BLK];
    __shared__ int   i1_s[TOK_PER_BLK];
    __shared__ int   i2_s[TOK_PER_BLK];

    const int tid  = threadIdx.x;
    const int lane = tid & 31;
    const int wv   = tid >> 5;                // wave 0..3, 16 tokens each
    const int blk  = blockIdx.x;
    const int tokBase = blk * TOK_PER_BLK + wv * 16;

    // WMMA fragment lane mapping (ISA 16x16x4 f32 layout)
    const int m     = lane & 15;              // A row / B col
    const int khalf = (lane >> 4) << 1;       // 0 or 2 (K half per lane group)
    const float* __restrict__ xrow = x + (size_t)(tokBase + m) * DM;

    // per-thread W-slab staging addresses (constant across slabs except kc)
    const int sidx = tid;                     // this thread covers idx = r*128+tid
    v8f acc0 = {}; v8f acc1 = {}; v8f acc2 = {}; v8f acc3 = {};

    for (int kc = 0; kc < DM; kc += KC) {
        __syncthreads();                      // previous slab fully consumed
        // ---- async Global->LDS stage of W[0:64][kc:kc+KC] (ASYNCcnt path) ----
        #pragma unroll
        for (int r = 0; r < 16; ++r) {
            int idx = r * 128 + sidx;         // 2048 float4 total
            int e   = idx >> 5;               // expert row
            int k4  = (idx & 31) << 2;        // k offset within slab (floats)
            unsigned     lds_b = (unsigned)(uintptr_t)(&smem[e * SW + k4]);
            const float* gsrc  = Wg + (size_t)e * DM + kc + k4;
            asm volatile("global_load_async_to_lds_b128 %0, %1, off"
                         :: "v"(lds_b), "v"(gsrc)
                         : "memory");
        }
        asm volatile("s_wait_asynccnt 0" ::: "memory");
        __syncthreads();                      // slab visible to all waves
        if (kc + KC < DM)
            __builtin_prefetch(xrow + kc + KC, 0, 0);   // global_prefetch next x slab
        #pragma unroll 2
        for (int kk = 0; kk < KC; kk += 4) {
            v2f a  = *(const v2f*)(xrow + kc + kk + khalf);
            v2f b0 = *(const v2f*)(&smem[( 0 + m) * SW + kk + khalf]);
            v2f b1 = *(const v2f*)(&smem[(16 + m) * SW + kk + khalf]);
            v2f b2 = *(const v2f*)(&smem[(32 + m) * SW + kk + khalf]);
            v2f b3 = *(const v2f*)(&smem[(48 + m) * SW + kk + khalf]);
            acc0 = __builtin_amdgcn_wmma_f32_16x16x4_f32(false, a, false, b0, (short)0, acc0, false, false);
            acc1 = __builtin_amdgcn_wmma_f32_16x16x4_f32(false, a, false, b1, (short)0, acc1, false, false);
            acc2 = __builtin_amdgcn_wmma_f32_16x16x4_f32(false, a, false, b2, (short)0, acc2, false, false);
            acc3 = __builtin_amdgcn_wmma_f32_16x16x4_f32(false, a, false, b3, (short)0, acc3, false, false);
        }
    }

    __syncthreads();                          // everyone done reading W slab
    // bias add + spill logits to LDS (C/D layout: VGPR v -> row v or v+8)
    {
        const int col  = lane & 15;
        const int rofs = (lane >> 4) << 3;    // 0 or 8
        v8f accs[4] = {acc0, acc1, acc2, acc3};
        #pragma unroll
        for (int j = 0; j < 4; ++j) {
            float bias = bg[j * 16 + col];
            #pragma unroll
            for (int v = 0; v < 8; ++v)
                smem[(wv * 16 + rofs + v) * SL + j * 16 + col] = accs[j][v] + bias;
        }
    }
    __syncthreads();

    // pass 1: one thread per token -> softmax stats + top2 (tie-break = lower index)
    if (tid < TOK_PER_BLK) {
        const float* lg = &smem[tid * SL];
        float mx = lg[0];
        #pragma unroll 8
        for (int e = 1; e < NEXP; ++e) mx = fmaxf(mx, lg[e]);
        float s = 0.f;
        #pragma unroll 8
        for (int e = 0; e < NEXP; ++e) s += __expf(lg[e] - mx);
        float inv = 1.0f / s;
        float lse = mx + __logf(s);
        float v1 = -__builtin_inff(), v2 = -__builtin_inff();
        int   i1 = 0, i2 = 0;
        for (int e = 0; e < NEXP; ++e) {
            float l = lg[e];
            if (l > v1)      { v2 = v1; i2 = i1; v1 = l; i1 = e; }
            else if (l > v2) { v2 = l;  i2 = e; }
        }
        mx_s[tid] = mx; inv_s[tid] = inv; lse2_s[tid] = lse * lse;
        i1_s[tid] = i1; i2_s[tid] = i2;
        int tg = blk * TOK_PER_BLK + tid;
        topi[tg] = make_int2(i1, i2);
        topp[tg] = make_float2(__expf(v1 - mx) * inv, __expf(v2 - mx) * inv);
    }
    __syncthreads();

    // pass 2: one thread per expert -> deterministic per-block reductions
    if (tid < NEXP) {
        const int e = tid;
        float ps = 0.f; int c = 0;
        for (int t = 0; t < TOK_PER_BLK; ++t) {
            ps += __expf(smem[t * SL + e] - mx_s[t]) * inv_s[t];
            c  += (i1_s[t] == e) + (i2_s[t] == e);
        }
        cnt_blk[blk * NEXP + e]  = c;
        psum_blk[blk * NEXP + e] = ps;
        if (e == 0) {
            float z = 0.f;
            for (int t = 0; t < TOK_PER_BLK; ++t) z += lse2_s[t];
            lse2_blk[blk] = z;
        }
    }
}

// =====================================================================
// Phase B: per-expert exclusive prefix over blocks + final loss scalar
// =====================================================================
__global__ __launch_bounds__(64)
void moe_phaseB(const int* __restrict__ cnt, int* __restrict__ offs,
                const float* __restrict__ psum, const float* __restrict__ lse2,
                float* __restrict__ loss_out)
{
    __shared__ float part[NEXP];
    const int e = threadIdx.x;
    int run = 0; float ps = 0.f;
    for (int b = 0; b < NBLK; ++b) {
        offs[b * NEXP + e] = run;
        run += cnt[b * NEXP + e];
        ps  += psum[b * NEXP + e];
    }
    const float invT = 1.0f / (float)T_TOK;
    part[e] = ((float)run * invT) * (ps * invT);
    __syncthreads();
    if (e == 0) {
        float aux = 0.f;
        for (int i = 0; i < NEXP; ++i) aux += part[i];
        aux *= ALPHA_C * (float)NEXP;
        float z = 0.f;
        for (int b = 0; b < NBLK; ++b) z += lse2[b];
        *loss_out = aux + z * invT;
    }
}

// =====================================================================
// Phase C: within-block ranks (exact token order) + dense output write
// =====================================================================
__global__ __launch_bounds__(TOK_PER_BLK)
void moe_phaseC(const int2* __restrict__ topi, const float2* __restrict__ topp,
                const int* __restrict__ offs,
                float* __restrict__ capped, float* __restrict__ weight)
{
    __shared__ int2 s_idx[TOK_PER_BLK];
    const int tid = threadIdx.x;
    const int blk = blockIdx.x;
    const int tg  = blk * TOK_PER_BLK + tid;

    int2   my = topi[tg];
    float2 mp = topp[tg];
    s_idx[tid] = my;
    __syncthreads();

    int r1 = 0, r2 = 0;
    for (int t = 0; t < tid; ++t) {
        int2 o = s_idx[t];
        r1 += (o.x == my.x) + (o.y == my.x);
        r2 += (o.x == my.y) + (o.y == my.y);
    }
    int pos1 = offs[blk * NEXP + my.x] + r1 + 1;   // inclusive cumsum position
    int pos2 = offs[blk * NEXP + my.y] + r2 + 1;
    bool k1 = pos1 < CAP, k2 = pos2 < CAP;

    float* cm = capped + (size_t)tg * NEXP;
    float* tw = weight + (size_t)tg * NEXP;
    float4 z4 = make_float4(0.f, 0.f, 0.f, 0.f);
    #pragma unroll
    for (int i = 0; i < 16; ++i) {
        *(float4*)(cm + 4 * i) = z4;
        *(float4*)(tw + 4 * i) = z4;
    }
    cm[my.x] = k1 ? 1.0f : 0.0f;
    cm[my.y] = k2 ? 1.0f : 0.0f;
    tw[my.x] = k1 ? mp.x : 0.0f;
    tw[my.y] = k2 ? mp.y : 0.0f;
}

// =====================================================================
extern "C" void kernel_launch(void* const* d_in, const int* in_sizes, int n_in,
                              void* d_out, int out_size, void* d_ws, size_t ws_size,
                              hipStream_t stream)
{
    const float* x  = (const float*)d_in[0];   // [4,4096,2048] f32
    const float* Wg = (const float*)d_in[1];   // [64,2048] f32
    const float* bg = (const float*)d_in[2];   // [64] f32

    char* ws = (char*)d_ws;
    int*    cnt  = (int*)   (ws + WS_CNT);
    int*    offs = (int*)   (ws + WS_OFFS);
    float*  psum = (float*) (ws + WS_PSUM);
    float*  lse2 = (float*) (ws + WS_LSE2);
    int2*   topi = (int2*)  (ws + WS_TOPI);
    float2* topp = (float2*)(ws + WS_TOPP);

    float* out    = (float*)d_out;
    float* capped = out;                                  // [T,64]
    float* weight = out + (size_t)T_TOK * NEXP;           // [T,64]
    float* loss   = out + 2 * (size_t)T_TOK * NEXP;       // scalar

    hipLaunchKernelGGL(moe_phaseA, dim3(NBLK), dim3(128), 0, stream,
                       x, Wg, bg, cnt, psum, lse2, topi, topp);
    hipLaunchKernelGGL(moe_phaseB, dim3(1), dim3(64), 0, stream,
                       cnt, offs, psum, lse2, loss);
    hipLaunchKernelGGL(moe_phaseC, dim3(NBLK), dim3(TOK_PER_BLK), 0, stream,
                       topi, topp, offs, capped, weight);
}